// PoseGCN_39247411151124
// MI455X (gfx1250) — compile-verified
//
#include <hip/hip_runtime.h>

// Problem constants (fixed by the reference)
#define NN   4096
#define EE   131072
#define GG   128
#define FIN  128
#define FH1  256
#define FH2  256
#define FOUT 128
#define NSWEEPS 8

typedef float v2f __attribute__((ext_vector_type(2)));
typedef float v8f __attribute__((ext_vector_type(8)));

// ======================= utility kernels =======================

__global__ void k_fill(float* __restrict__ p, size_t n, float v) {
  size_t i = (size_t)blockIdx.x * blockDim.x + threadIdx.x;
  size_t st = (size_t)gridDim.x * blockDim.x;
  for (; i < n; i += st) p[i] = v;
}

__global__ void k_scatter_adj(const int* __restrict__ ei, float* __restrict__ A) {
  int e = blockIdx.x * blockDim.x + threadIdx.x;
  if (e < EE) A[(size_t)ei[e] * NN + ei[EE + e]] = 1.0f;  // duplicates collapse to 1
}

__global__ void k_row_deg(const float* __restrict__ A, float* __restrict__ deg) {
  __shared__ float red[256];
  int row = blockIdx.x;
  float s = 0.f;
  for (int j = threadIdx.x; j < NN; j += 256) s += A[(size_t)row * NN + j];
  red[threadIdx.x] = s;
  __syncthreads();
  for (int k = 128; k > 0; k >>= 1) {
    if ((int)threadIdx.x < k) red[threadIdx.x] += red[threadIdx.x + k];
    __syncthreads();
  }
  if (threadIdx.x == 0) deg[row] = red[0];
}

__global__ void k_make_lap(float* __restrict__ A, const float* __restrict__ deg) {
  size_t n = (size_t)NN * NN;
  size_t i = (size_t)blockIdx.x * blockDim.x + threadIdx.x;
  size_t st = (size_t)gridDim.x * blockDim.x;
  for (; i < n; i += st) {
    size_t r = i / NN, c = i % NN;
    A[i] = ((r == c) ? deg[r] : 0.0f) - A[i];
  }
}

__global__ void k_init_identity(float* __restrict__ U) {
  size_t n = (size_t)NN * NN;
  size_t i = (size_t)blockIdx.x * blockDim.x + threadIdx.x;
  size_t st = (size_t)gridDim.x * blockDim.x;
  for (; i < n; i += st) {
    size_t r = i / NN, c = i % NN;
    U[i] = (r == c) ? 1.0f : 0.0f;
  }
}

// GCN normalization: deg (with self-loops) -> d^{-1/2}
__global__ void k_count_col(const int* __restrict__ ei, float* __restrict__ d) {
  int e = blockIdx.x * blockDim.x + threadIdx.x;
  if (e < EE) atomicAdd(&d[ei[EE + e]], 1.0f);
}

__global__ void k_finish_dinv(float* __restrict__ d) {
  int i = blockIdx.x * blockDim.x + threadIdx.x;
  if (i < NN) d[i] = rsqrtf(d[i]);  // always >= 1 due to self-loop
}

// ======================= Jacobi eigensolver =======================
// Parallel cyclic Jacobi: per round, 2048 disjoint (p,q) pairs from the
// round-robin tournament schedule.  A and U (128 MB) stay resident in the
// 192 MB L2, so the O(N^3) rotation traffic never touches HBM.

__device__ __forceinline__ void jpair(int rnd, int i, int& p, int& q) {
  const int m = NN - 1;
  int a = (rnd + i) % m;
  int b = (i == 0) ? m : ((rnd + m - i) % m);
  p = a < b ? a : b;
  q = a < b ? b : a;
}

__global__ void k_jacobi_angles(const float* __restrict__ A, int rnd,
                                float* __restrict__ cs) {
  int i = blockIdx.x * blockDim.x + threadIdx.x;
  if (i >= NN / 2) return;
  int p, q; jpair(rnd, i, p, q);
  float app = A[(size_t)p * NN + p];
  float aqq = A[(size_t)q * NN + q];
  float apq = A[(size_t)p * NN + q];
  float c = 1.0f, s = 0.0f;
  if (fabsf(apq) > 1e-30f) {
    float tau = (aqq - app) / (2.0f * apq);
    float t = copysignf(1.0f, tau) / (fabsf(tau) + sqrtf(1.0f + tau * tau));
    c = rsqrtf(1.0f + t * t);
    s = t * c;
  }
  cs[2 * i] = c;
  cs[2 * i + 1] = s;
}

// Row rotation: rows are contiguous -> float4 (global_load_b128 / store_b128).
__global__ void k_jacobi_rowrot(float* __restrict__ A, const float* __restrict__ cs,
                                int rnd) {
  int i = blockIdx.y;                                        // pair index (0..2047)
  int j4 = (blockIdx.x * blockDim.x + threadIdx.x) * 4;      // column quad
  int p, q; jpair(rnd, i, p, q);
  float c = cs[2 * i], s = cs[2 * i + 1];
  float4* Ap = (float4*)&A[(size_t)p * NN + j4];
  float4* Aq = (float4*)&A[(size_t)q * NN + j4];
  float4 ap = *Ap, aq = *Aq, np, nq;
  np.x = c * ap.x - s * aq.x;  nq.x = s * ap.x + c * aq.x;
  np.y = c * ap.y - s * aq.y;  nq.y = s * ap.y + c * aq.y;
  np.z = c * ap.z - s * aq.z;  nq.z = s * ap.z + c * aq.z;
  np.w = c * ap.w - s * aq.w;  nq.w = s * ap.w + c * aq.w;
  *Ap = np; *Aq = nq;
}

// Column rotation: stride-N accesses; 4 rows per thread for MLP (L2 latency hiding).
__global__ void k_jacobi_colrot(float* __restrict__ A, float* __restrict__ U,
                                const float* __restrict__ cs, int rnd) {
  int i = blockIdx.y;                                        // pair index
  int j0 = (blockIdx.x * blockDim.x + threadIdx.x) * 4;      // row base
  int p, q; jpair(rnd, i, p, q);
  float c = cs[2 * i], s = cs[2 * i + 1];
#pragma unroll
  for (int r = 0; r < 4; ++r) {
    size_t ip = (size_t)(j0 + r) * NN + p, iq = (size_t)(j0 + r) * NN + q;
    float ap = A[ip], aq = A[iq];
    A[ip] = c * ap - s * aq;
    A[iq] = s * ap + c * aq;
    float up = U[ip], uq = U[iq];
    U[ip] = c * up - s * uq;
    U[iq] = s * up + c * uq;
  }
}

// ======================= WMMA f32 GEMM (async-LDS double buffered) ==========
// C = op(A) @ B ; f32 in, f32 accumulate via V_WMMA_F32_16X16X4_F32.
// Tiles are streamed memory->LDS with GLOBAL_LOAD_ASYNC_TO_LDS_B32 (ASYNCcnt),
// double-buffered: while computing tile k, tile k+1 is in flight.
// 256 threads = 8 waves; waves tiled 4x2, each wave owns one 16x16 C tile.

#define BM 64
#define BN 32
#define BK 32
#define LDA_P (BK + 1)                       // padded As row stride (floats)
#define LDB_P (BN + 1)                       // padded Bs row stride (floats)
#define AS_FLOATS (BM * LDA_P)               // 2112
#define BS_FLOATS (BK * LDB_P)               // 1056
#define BUF_FLOATS (AS_FLOATS + BS_FLOATS)   // 3168
#define GEMM_SMEM_BYTES (2 * BUF_FLOATS * 4) // 25344 B dynamic LDS
#define LOADS_PER_TILE 12                    // 8 (A) + 4 (B) b32 loads per thread

__device__ __forceinline__ void async_ld_b32(unsigned lds_off, const float* g) {
  asm volatile("global_load_async_to_lds_b32 %0, %1, off"
               :: "v"(lds_off), "v"((unsigned long long)(uintptr_t)g)
               : "memory");
}

__launch_bounds__(256)
__global__ void k_gemm_wmma(const float* __restrict__ A, const float* __restrict__ B,
                            float* __restrict__ C, int M, int Nc, int K, int transA) {
  extern __shared__ float smem[];
  const int tid  = threadIdx.x;
  const int lane = tid & 31;
  const int wave = tid >> 5;
  const int wm = wave & 3;          // wave tile row (0..3)
  const int wn = wave >> 2;         // wave tile col (0..1)
  const int m0 = blockIdx.x * BM;
  const int n0 = blockIdx.y * BN;

  // Wave-relative LDS byte offset of the dynamic-LDS base (low 32 bits of the
  // generic address of an LDS pointer are the ds-space byte offset).
  const unsigned ldsBase = (unsigned)(uintptr_t)(void*)smem;

  auto issue_tile = [&](int k0, int b) {
    const unsigned aOff = ldsBase + (unsigned)(b * BUF_FLOATS) * 4u;
#pragma unroll
    for (int c = 0, i = tid; c < 8; ++c, i += 256) {          // A: 64x32
      int m = i / BK, k = i % BK;
      const float* g = transA ? &A[(size_t)(k0 + k) * M + (m0 + m)]
                              : &A[(size_t)(m0 + m) * K + (k0 + k)];
      async_ld_b32(aOff + (unsigned)(m * LDA_P + k) * 4u, g);
    }
    const unsigned bOff = aOff + (unsigned)AS_FLOATS * 4u;
#pragma unroll
    for (int c = 0, i = tid; c < 4; ++c, i += 256) {          // B: 32x32
      int k = i / BN, n = i % BN;
      async_ld_b32(bOff + (unsigned)(k * LDB_P + n) * 4u,
                   &B[(size_t)(k0 + k) * Nc + (n0 + n)]);
    }
  };

  v8f acc = {0.f, 0.f, 0.f, 0.f, 0.f, 0.f, 0.f, 0.f};

  issue_tile(0, 0);                 // prologue: tile 0 in flight
  int cur = 0;
  for (int k0 = 0; k0 < K; k0 += BK) {
    const bool more = (k0 + BK) < K;
    if (more) issue_tile(k0 + BK, cur ^ 1);   // overlap next tile with compute
    // Async loads complete in order: waiting down to LOADS_PER_TILE outstanding
    // guarantees the current tile has landed while the next stays in flight.
    if (more) asm volatile("s_wait_asynccnt 0xc" ::: "memory");
    else      asm volatile("s_wait_asynccnt 0x0" ::: "memory");
    __syncthreads();                // make all waves' async LDS writes visible

    const float* As = smem + cur * BUF_FLOATS;
    const float* Bs = As + AS_FLOATS;
    const int half = lane >> 4;     // lanes 16-31 hold K+2..K+3 of the fragment
    const int l = lane & 15;
#pragma unroll
    for (int kk = 0; kk < BK; kk += 4) {
      v2f a, b;
      a.x = As[(wm * 16 + l) * LDA_P + kk + half * 2 + 0];
      a.y = As[(wm * 16 + l) * LDA_P + kk + half * 2 + 1];
      b.x = Bs[(kk + half * 2 + 0) * LDB_P + wn * 16 + l];
      b.y = Bs[(kk + half * 2 + 1) * LDB_P + wn * 16 + l];
      acc = __builtin_amdgcn_wmma_f32_16x16x4_f32(false, a, false, b,
                                                  (short)0, acc, false, false);
    }
    __syncthreads();                // all waves done reading before overwrite
    cur ^= 1;
  }

  // C/D layout: VGPR r -> M = r (lanes 0-15) or M = r+8 (lanes 16-31), N = lane&15
  const int l = lane & 15;
  const int rowbase = m0 + wm * 16 + ((lane >> 4) ? 8 : 0);
  const int col = n0 + wn * 16 + l;
#pragma unroll
  for (int r = 0; r < 8; ++r)
    C[(size_t)(rowbase + r) * Nc + col] = acc[r];
}

// ======================= GCN aggregation / bias / pooling =======================

__global__ void k_message(const int* __restrict__ ei, const float* __restrict__ dinv,
                          const float* __restrict__ hin, float* __restrict__ hout,
                          int F) {
  size_t total = (size_t)(EE + NN) * F;
  size_t i = (size_t)blockIdx.x * blockDim.x + threadIdx.x;
  size_t st = (size_t)gridDim.x * blockDim.x;
  for (; i < total; i += st) {
    int e = (int)(i / F);
    int f = (int)(i % F);
    int r, c;
    if (e < EE) { r = ei[e]; c = ei[EE + e]; }
    else        { r = c = e - EE; }            // self-loop
    float w = dinv[r] * dinv[c];
    atomicAdd(&hout[(size_t)c * F + f], hin[(size_t)r * F + f] * w);
  }
}

__global__ void k_bias_act(float* __restrict__ h, const float* __restrict__ b,
                           int F, int relu) {
  size_t total = (size_t)NN * F;
  size_t i = (size_t)blockIdx.x * blockDim.x + threadIdx.x;
  size_t st = (size_t)gridDim.x * blockDim.x;
  for (; i < total; i += st) {
    float v = h[i] + b[i % F];
    h[i] = relu ? fmaxf(v, 0.0f) : v;
  }
}

__global__ void k_pool(const float* __restrict__ h, const int* __restrict__ batch,
                       float* __restrict__ sums, float* __restrict__ cnts) {
  size_t total = (size_t)NN * FOUT;
  size_t i = (size_t)blockIdx.x * blockDim.x + threadIdx.x;
  size_t st = (size_t)gridDim.x * blockDim.x;
  for (; i < total; i += st) {
    int node = (int)(i / FOUT);
    int f = (int)(i % FOUT);
    int g = batch[node];
    atomicAdd(&sums[(size_t)g * FOUT + f], h[i]);
    if (f == 0) atomicAdd(&cnts[g], 1.0f);
  }
}

__global__ void k_pool_div(const float* __restrict__ sums, const float* __restrict__ cnts,
                           float* __restrict__ out) {
  int i = blockIdx.x * blockDim.x + threadIdx.x;
  if (i < GG * FOUT) out[i] = sums[i] / fmaxf(cnts[i / FOUT], 1.0f);
}

// ======================= launch =======================

extern "C" void kernel_launch(void* const* d_in, const int* in_sizes, int n_in,
                              void* d_out, int out_size, void* d_ws, size_t ws_size,
                              hipStream_t stream) {
  const float* x     = (const float*)d_in[0];
  const int*   ei    = (const int*)d_in[1];
  const int*   batch = (const int*)d_in[2];
  // d_in[3] = num_graphs (fixed 128)
  const float* W1 = (const float*)d_in[4];
  const float* b1 = (const float*)d_in[5];
  const float* W2 = (const float*)d_in[6];
  const float* b2 = (const float*)d_in[7];
  const float* W3 = (const float*)d_in[8];
  const float* b3 = (const float*)d_in[9];
  float* out = (float*)d_out;

  // workspace layout (floats)
  float* A    = (float*)d_ws;                         // 4096*4096
  float* U    = A  + (size_t)NN * NN;                 // 4096*4096
  float* hA   = U  + (size_t)NN * NN;                 // 4096*256
  float* hB   = hA + (size_t)NN * 256;                // 4096*256
  float* hC   = hB + (size_t)NN * 256;                // 4096*256
  float* dinv = hC + (size_t)NN * 256;                // 4096
  float* scr  = dinv + NN;                            // 4096 (deg, then c/s pairs)
  float* sums = scr + NN;                             // 128*128
  float* cnts = sums + (size_t)GG * FOUT;             // 128

  const dim3 T(256);
  const dim3 GBIG(8192);                  // grid-stride fills / elementwise
  const dim3 GROT(NN / (256 * 4), NN / 2);  // 4-wide rotation kernels

  // ---- Laplacian ----
  k_fill<<<GBIG, T, 0, stream>>>(A, (size_t)NN * NN, 0.0f);
  k_scatter_adj<<<EE / 256, T, 0, stream>>>(ei, A);
  k_row_deg<<<NN, T, 0, stream>>>(A, scr);
  k_make_lap<<<GBIG, T, 0, stream>>>(A, scr);
  k_init_identity<<<GBIG, T, 0, stream>>>(U);

  // ---- GCN degree normalization (with self-loops) ----
  k_fill<<<NN / 256, T, 0, stream>>>(dinv, NN, 1.0f);
  k_count_col<<<EE / 256, T, 0, stream>>>(ei, dinv);
  k_finish_dinv<<<NN / 256, T, 0, stream>>>(dinv);

  // ---- eigh via parallel cyclic Jacobi (A,U stay L2-resident) ----
  for (int sweep = 0; sweep < NSWEEPS; ++sweep) {
    for (int rnd = 0; rnd < NN - 1; ++rnd) {
      k_jacobi_angles<<<(NN / 2) / 256, T, 0, stream>>>(A, rnd, scr);
      k_jacobi_rowrot<<<GROT, T, 0, stream>>>(A, scr, rnd);
      k_jacobi_colrot<<<GROT, T, 0, stream>>>(A, U, scr, rnd);
    }
  }

  // ---- GFT: hA = U^T @ x  (4096x128) ----
  k_gemm_wmma<<<dim3(NN / BM, FIN / BN), T, GEMM_SMEM_BYTES, stream>>>(
      U, x, hA, NN, FIN, NN, 1);

  // ---- layer 1: hC = relu(agg(hA @ W1) + b1) ----
  k_gemm_wmma<<<dim3(NN / BM, FH1 / BN), T, GEMM_SMEM_BYTES, stream>>>(
      hA, W1, hB, NN, FH1, FIN, 0);
  k_fill<<<GBIG, T, 0, stream>>>(hC, (size_t)NN * FH1, 0.0f);
  k_message<<<GBIG, T, 0, stream>>>(ei, dinv, hB, hC, FH1);
  k_bias_act<<<GBIG, T, 0, stream>>>(hC, b1, FH1, 1);

  // ---- layer 2: hA = relu(agg(hC @ W2) + b2) ----
  k_gemm_wmma<<<dim3(NN / BM, FH2 / BN), T, GEMM_SMEM_BYTES, stream>>>(
      hC, W2, hB, NN, FH2, FH1, 0);
  k_fill<<<GBIG, T, 0, stream>>>(hA, (size_t)NN * FH2, 0.0f);
  k_message<<<GBIG, T, 0, stream>>>(ei, dinv, hB, hA, FH2);
  k_bias_act<<<GBIG, T, 0, stream>>>(hA, b2, FH2, 1);

  // ---- layer 3: hC = agg(hA @ W3) + b3 ----
  k_gemm_wmma<<<dim3(NN / BM, FOUT / BN), T, GEMM_SMEM_BYTES, stream>>>(
      hA, W3, hB, NN, FOUT, FH2, 0);
  k_fill<<<GBIG, T, 0, stream>>>(hC, (size_t)NN * FOUT, 0.0f);
  k_message<<<GBIG, T, 0, stream>>>(ei, dinv, hB, hC, FOUT);
  k_bias_act<<<GBIG, T, 0, stream>>>(hC, b3, FOUT, 0);

  // ---- inverse GFT: hA = U @ hC ----
  k_gemm_wmma<<<dim3(NN / BM, FOUT / BN), T, GEMM_SMEM_BYTES, stream>>>(
      U, hC, hA, NN, FOUT, NN, 0);

  // ---- global mean pool ----
  k_fill<<<(GG * FOUT) / 256, T, 0, stream>>>(sums, (size_t)GG * FOUT, 0.0f);
  k_fill<<<1, T, 0, stream>>>(cnts, (size_t)GG, 0.0f);
  k_pool<<<GBIG, T, 0, stream>>>(hA, batch, sums, cnts);
  k_pool_div<<<(GG * FOUT) / 256, T, 0, stream>>>(sums, cnts, out);
}